// GCNConvLearnedVN_77506979823737
// MI455X (gfx1250) — compile-verified
//
#include <hip/hip_runtime.h>
#include <cmath>

#define N_NODES 50000
#define N_EDGES 800000
#define DIM 128

typedef float v2f __attribute__((ext_vector_type(2)));
typedef float v8f __attribute__((ext_vector_type(8)));

// ---------------------------------------------------------------------------
// 1) Per-node partial dot products for the edge-weight predictor:
//    p_a[n] = x[n] . w_pred[0:128],  p_b[n] = x[n] . w_pred[128:256]
//    Also seeds deg[n] = 1.0 (self-loop weight).
//    One wave32 per node; 4 elements per lane; shfl reduction.
// ---------------------------------------------------------------------------
__global__ __launch_bounds__(256) void node_dots_kernel(
    const float* __restrict__ x, const float* __restrict__ w_pred,
    float* __restrict__ p_a, float* __restrict__ p_b, float* __restrict__ deg)
{
    int node = (blockIdx.x * blockDim.x + threadIdx.x) >> 5;
    int lane = threadIdx.x & 31;
    if (node >= N_NODES) return;
    const float* xr = x + (size_t)node * DIM;
    float sa = 0.f, sb = 0.f;
#pragma unroll
    for (int j = 0; j < 4; ++j) {
        int c = lane + 32 * j;
        float v = xr[c];
        sa += v * w_pred[c];
        sb += v * w_pred[DIM + c];
    }
#pragma unroll
    for (int off = 16; off > 0; off >>= 1) {
        sa += __shfl_down(sa, off, 32);
        sb += __shfl_down(sb, off, 32);
    }
    if (lane == 0) {
        p_a[node] = sa;
        p_b[node] = sb;
        deg[node] = 1.0f;  // self-loop contribution to weighted in-degree
    }
}

// ---------------------------------------------------------------------------
// 2) Per-edge learned weight + weighted-degree accumulation.
// ---------------------------------------------------------------------------
__global__ __launch_bounds__(256) void edge_weight_kernel(
    const int* __restrict__ src, const int* __restrict__ tgt,
    const float* __restrict__ p_a, const float* __restrict__ p_b,
    const float* __restrict__ b_pred,
    float* __restrict__ ew, float* __restrict__ deg)
{
    int e = blockIdx.x * blockDim.x + threadIdx.x;
    if (e >= N_EDGES) return;
    int s = src[e], t = tgt[e];
    float z = p_a[s] + p_b[t] + b_pred[0];
    float w = 1.0f / (1.0f + __expf(-z));
    ew[e] = w;
    unsafeAtomicAdd(&deg[t], w);   // hardware global_atomic_add_f32
}

// ---------------------------------------------------------------------------
// 3) deg -> deg^{-1/2} in place (deg >= 1 always, guard anyway)
// ---------------------------------------------------------------------------
__global__ __launch_bounds__(256) void rsqrt_kernel(float* __restrict__ deg_dis)
{
    int n = blockIdx.x * blockDim.x + threadIdx.x;
    if (n < N_NODES) {
        float d = deg_dis[n];
        deg_dis[n] = (d > 0.f) ? rsqrtf(d) : 0.f;
    }
}

// ---------------------------------------------------------------------------
// 4) xw = x @ W with V_WMMA_F32_16X16X4_F32.
//    Block = 256 threads (8 waves). Block handles a 16-row stripe; wave w
//    handles columns [16w, 16w+16). A tile (16x128) staged in LDS with
//    row stride 132 to avoid 16-way bank conflicts.
//    A layout (16x4 f32): v0 = A[m][k0+2*hi], v1 = A[m][k0+2*hi+1]
//    B layout (4x16 f32): v0 = B[k0+2*hi][n], v1 = B[k0+2*hi+1][n]
//    D layout: VGPR i -> row (i + 8*hi), col n  (m = n = lane & 15, hi = lane>>4)
// ---------------------------------------------------------------------------
__global__ __launch_bounds__(256) void gemm_wmma_kernel(
    const float* __restrict__ x, const float* __restrict__ W,
    float* __restrict__ xw)
{
    __shared__ float As[16 * 132];
    const int rowBase = blockIdx.x * 16;

    // cooperative coalesced stage of the 16x128 A tile
#pragma unroll
    for (int j = 0; j < 8; ++j) {
        int idx = threadIdx.x + j * 256;
        int r = idx >> 7, c = idx & 127;
        As[r * 132 + c] = x[(size_t)rowBase * DIM + idx];
    }
    __syncthreads();

    const int waveId  = threadIdx.x >> 5;
    const int lane    = threadIdx.x & 31;
    const int m       = lane & 15;
    const int hi      = lane >> 4;        // 0: lanes 0-15, 1: lanes 16-31
    const int colBase = waveId * 16;

    v8f acc = {};
#pragma unroll 8
    for (int k0 = 0; k0 < DIM; k0 += 4) {
        int ka = k0 + 2 * hi;
        v2f a, b;
        a.x = As[m * 132 + ka];
        a.y = As[m * 132 + ka + 1];
        b.x = W[(size_t)ka * DIM + colBase + m];
        b.y = W[(size_t)(ka + 1) * DIM + colBase + m];
        acc = __builtin_amdgcn_wmma_f32_16x16x4_f32(
            false, a, false, b, (short)0, acc, false, false);
    }

#pragma unroll
    for (int i = 0; i < 8; ++i) {
        int row = rowBase + i + 8 * hi;
        xw[(size_t)row * DIM + colBase + m] = acc[i];
    }
}

// ---------------------------------------------------------------------------
// 5) Self-loop term + bias; fully initializes d_out:
//    out[n][d] = dis[n]^2 * xw[n][d] + b[d]
// ---------------------------------------------------------------------------
__global__ __launch_bounds__(256) void self_bias_kernel(
    const float* __restrict__ xw, const float* __restrict__ dis,
    const float* __restrict__ b, float* __restrict__ out)
{
    int idx = blockIdx.x * blockDim.x + threadIdx.x;
    if (idx >= N_NODES * DIM) return;
    int node = idx >> 7, d = idx & 127;
    float s = dis[node];
    out[idx] = s * s * xw[idx] + b[d];
}

// ---------------------------------------------------------------------------
// 6) Edge scatter: one wave per edge; lane handles 4 consecutive features.
//    out[tgt] += dis[src]*ew*dis[tgt] * xw[src]  via hardware f32 atomics.
//    xw/out live in L2 (25.6 MB each, 192 MB L2) -> L2-rate scatter.
// ---------------------------------------------------------------------------
__global__ __launch_bounds__(256) void edge_scatter_kernel(
    const int* __restrict__ src, const int* __restrict__ tgt,
    const float* __restrict__ ew, const float* __restrict__ dis,
    const float* __restrict__ xw, float* __restrict__ out)
{
    int e = (blockIdx.x * blockDim.x + threadIdx.x) >> 5;
    int lane = threadIdx.x & 31;
    if (e >= N_EDGES) return;
    int s = src[e], t = tgt[e];
    float norm = dis[s] * ew[e] * dis[t];
    const float4* xv = (const float4*)(xw + (size_t)s * DIM);
    float4 v = xv[lane];
    float* op = out + (size_t)t * DIM + lane * 4;
    unsafeAtomicAdd(op + 0, norm * v.x);
    unsafeAtomicAdd(op + 1, norm * v.y);
    unsafeAtomicAdd(op + 2, norm * v.z);
    unsafeAtomicAdd(op + 3, norm * v.w);
}

// ---------------------------------------------------------------------------
extern "C" void kernel_launch(void* const* d_in, const int* in_sizes, int n_in,
                              void* d_out, int out_size, void* d_ws, size_t ws_size,
                              hipStream_t stream) {
    const float* x          = (const float*)d_in[0];
    const int*   edge_index = (const int*)d_in[1];
    const float* W          = (const float*)d_in[2];
    const float* b          = (const float*)d_in[3];
    const float* w_pred     = (const float*)d_in[4];
    const float* b_pred     = (const float*)d_in[5];
    float*       out        = (float*)d_out;

    const int* src = edge_index;            // row 0 of [2, E]
    const int* tgt = edge_index + N_EDGES;  // row 1 of [2, E]

    // workspace partition (floats): p_a[N], p_b[N], deg/dis[N], ew[E], xw[N*128]
    float* ws  = (float*)d_ws;
    float* p_a = ws;
    float* p_b = p_a + N_NODES;
    float* deg = p_b + N_NODES;   // becomes dis in place after rsqrt_kernel
    float* ew  = deg + N_NODES;
    float* xw  = ew + N_EDGES;

    node_dots_kernel  <<<(N_NODES + 7) / 8,      256, 0, stream>>>(x, w_pred, p_a, p_b, deg);
    edge_weight_kernel<<<(N_EDGES + 255) / 256,  256, 0, stream>>>(src, tgt, p_a, p_b, b_pred, ew, deg);
    rsqrt_kernel      <<<(N_NODES + 255) / 256,  256, 0, stream>>>(deg);
    gemm_wmma_kernel  <<<N_NODES / 16,           256, 0, stream>>>(x, W, xw);
    self_bias_kernel  <<<(N_NODES * DIM) / 256,  256, 0, stream>>>(xw, deg, b, out);
    edge_scatter_kernel<<<N_EDGES / 8,           256, 0, stream>>>(src, tgt, ew, deg, xw, out);
}